// MeanFieldGaussianFeedForward_69827578298631
// MI455X (gfx1250) — compile-verified
//
#include <hip/hip_runtime.h>
#include <math.h>

typedef __attribute__((ext_vector_type(2))) float v2f;
typedef __attribute__((ext_vector_type(4))) float v4f;
typedef __attribute__((ext_vector_type(8))) float v8f;

#define IN_DIM  1024
#define OUT_DIM 1024
#define B_DIM   1024

#define TILE_M 64
#define TILE_N 128
#define TILE_K 32
// LDS row strides padded for bank-conflict-free WMMA fragment reads
// (64 banks x 4B). 36: 36*r mod 64 distinct for r=0..15, and cross-half
// (+2 column) never aliases (36*d mod 64 is a multiple of 4, never 2).
// 136: 136 mod 64 == 8, so rows k and k+2 land 16 banks apart ->
// the two half-waves (16 consecutive columns each) hit disjoint banks.
#define A_STRIDE 36
#define B_STRIDE 136

__device__ __forceinline__ float softplus_f(float x) {
  // numerically stable: max(x,0) + log1p(exp(-|x|))
  return fmaxf(x, 0.0f) + log1pf(expf(-fabsf(x)));
}

// ---------------------------------------------------------------------------
// Kernel 1: W_eff = softplus(w_s) * mean_k(eps_k) + w_m
// Streams 419 MB of eps once (non-temporal: don't evict L2-resident GEMM
// operands), fully coalesced 16B traffic. This is the roofline floor
// (~18 us @ 23.3 TB/s) of the entire problem.
// ---------------------------------------------------------------------------
__global__ __launch_bounds__(256) void build_weff_kernel(
    const float* __restrict__ eps, const float* __restrict__ wm,
    const float* __restrict__ ws, float* __restrict__ weff,
    int K, float invK) {
  const int p = blockIdx.x * blockDim.x + threadIdx.x;  // v4f index
  const int n4 = (IN_DIM * OUT_DIM) / 4;
  if (p >= n4) return;

  const v4f* e4 = (const v4f*)eps;
  v4f acc = (v4f){0.f, 0.f, 0.f, 0.f};
#pragma unroll 4
  for (int k = 0; k < K; ++k) {
    v4f v = __builtin_nontemporal_load(&e4[(size_t)k * n4 + p]);
    acc += v;
  }

  const v4f m = ((const v4f*)wm)[p];
  const v4f s = ((const v4f*)ws)[p];
  v4f w;
  w.x = fmaf(softplus_f(s.x), acc.x * invK, m.x);
  w.y = fmaf(softplus_f(s.y), acc.y * invK, m.y);
  w.z = fmaf(softplus_f(s.z), acc.z * invK, m.z);
  w.w = fmaf(softplus_f(s.w), acc.w * invK, m.w);
  ((v4f*)weff)[p] = w;
}

// ---------------------------------------------------------------------------
// Kernel 2: out[B, OUT] = x[B, IN] @ W_eff[IN, OUT], full f32 via
// V_WMMA_F32_16X16X4_F32. 256 threads = 8 wave32; block tile 64x128
// (128 blocks -> fills the machine), waves arranged 4(M) x 2(N), each wave
// computes a 16x64 strip = 4 accumulator tiles (32 VGPRs of accum).
// ---------------------------------------------------------------------------
__global__ __launch_bounds__(256) void gemm_wmma_f32_kernel(
    const float* __restrict__ A,   // x        [B_DIM  x IN_DIM]
    const float* __restrict__ Bw,  // W_eff    [IN_DIM x OUT_DIM]
    float* __restrict__ C) {       // out      [B_DIM  x OUT_DIM]
  __shared__ float sA[TILE_M][A_STRIDE];  // 64x36 f32  ~ 9.2 KB
  __shared__ float sB[TILE_K][B_STRIDE];  // 32x136 f32 ~ 17.4 KB

  const int tid  = threadIdx.x;
  const int lane = tid & 31;        // wave32
  const int wave = tid >> 5;        // 0..7
  const int waveM = wave >> 1;      // 0..3 : 16-row strip
  const int waveN = wave & 1;       // 0..1 : 64-col half
  const int hi   = lane >> 4;       // half-wave select (K/M-split per ISA)
  const int l16  = lane & 15;

  const int m0 = blockIdx.y * TILE_M;
  const int n0 = blockIdx.x * TILE_N;
  const int aRow = waveM * 16 + l16;  // this lane's A-matrix row in the tile

  v8f acc[4] = {};  // 4 tiles x 8 VGPR f32 accumulators

  for (int k0 = 0; k0 < IN_DIM; k0 += TILE_K) {
    // --- stage A tile (64x32) : 512 x 16B, 2 per thread ---
#pragma unroll
    for (int i = tid; i < (TILE_M * TILE_K) / 4; i += 256) {
      const int r  = i >> 3;       // 8 v4f per row
      const int c4 = (i & 7) * 4;
      const v4f v = *(const v4f*)&A[(size_t)(m0 + r) * IN_DIM + k0 + c4];
      *(v4f*)&sA[r][c4] = v;
    }
    // --- stage B tile (32x128) : 1024 x 16B, 4 per thread ---
#pragma unroll
    for (int i = tid; i < (TILE_K * TILE_N) / 4; i += 256) {
      const int r  = i >> 5;       // 32 v4f per row
      const int c4 = (i & 31) * 4;
      const v4f v = *(const v4f*)&Bw[(size_t)(k0 + r) * OUT_DIM + n0 + c4];
      *(v4f*)&sB[r][c4] = v;
    }
    __syncthreads();

#pragma unroll
    for (int kk = 0; kk < TILE_K; kk += 4) {
      // A fragment 16x4 f32: v0 = K=(0|2), v1 = K=(1|3) per half-wave
      v2f a;
      a.x = sA[aRow][kk + hi * 2 + 0];
      a.y = sA[aRow][kk + hi * 2 + 1];
#pragma unroll
      for (int t = 0; t < 4; ++t) {
        // B fragment 4x16 f32: row striped across lanes within a VGPR
        const int col = waveN * 64 + t * 16 + l16;
        v2f b;
        b.x = sB[kk + hi * 2 + 0][col];
        b.y = sB[kk + hi * 2 + 1][col];
        // 8 args: (neg_a, A, neg_b, B, c_mod, C, reuse_a, reuse_b)
        acc[t] = __builtin_amdgcn_wmma_f32_16x16x4_f32(
            false, a, false, b, (short)0, acc[t], false, false);
      }
    }
    __syncthreads();
  }

  // D layout: VGPR r holds row M=r (lanes 0-15) / M=r+8 (lanes 16-31)
#pragma unroll
  for (int t = 0; t < 4; ++t) {
    const int col = n0 + waveN * 64 + t * 16 + l16;
#pragma unroll
    for (int r = 0; r < 8; ++r) {
      const int row = m0 + waveM * 16 + hi * 8 + r;
      C[(size_t)row * OUT_DIM + col] = acc[t][r];
    }
  }
}

extern "C" void kernel_launch(void* const* d_in, const int* in_sizes, int n_in,
                              void* d_out, int out_size, void* d_ws, size_t ws_size,
                              hipStream_t stream) {
  const float* x   = (const float*)d_in[0];  // [1024, 1024]
  const float* eps = (const float*)d_in[1];  // [K, 1024, 1024]
  const float* wm  = (const float*)d_in[2];  // [1024, 1024]
  const float* ws  = (const float*)d_in[3];  // [1024, 1024]
  float* out  = (float*)d_out;               // [1024, 1024]
  float* weff = (float*)d_ws;                // 4 MB scratch: W_eff

  const int K = in_sizes[1] / (IN_DIM * OUT_DIM);  // 100

  // Kernel 1: fused eps-mean + softplus reparameterization -> W_eff
  {
    const int n4 = (IN_DIM * OUT_DIM) / 4;     // 262144 v4f
    dim3 grid((n4 + 255) / 256), block(256);
    build_weff_kernel<<<grid, block, 0, stream>>>(eps, wm, ws, weff, K,
                                                  1.0f / (float)K);
  }
  // Kernel 2: out = x @ W_eff (f32 WMMA)
  {
    dim3 grid(OUT_DIM / TILE_N, B_DIM / TILE_M), block(256);
    gemm_wmma_f32_kernel<<<grid, block, 0, stream>>>(x, weff, out);
  }
}